// Weights_6622839570439
// MI455X (gfx1250) — compile-verified
//
#include <hip/hip_runtime.h>

// Elementwise pacing-weight kernel for MI455X (gfx1250).
// w_i = (loss_i > eta) ? 0 : 1 - loss_i/eta,  eta = clamp(eta_value[0], ln2, 10)
// Pure streaming op: 256 MB traffic @ 23.3 TB/s -> bandwidth bound (~11 us floor).
// Strategy: b128 vector loads/stores with non-temporal cache policy (working set
// > 192 MB L2, no reuse), uniform eta hoisted to scalar path, 1 FMA/element.

typedef __attribute__((ext_vector_type(4))) float v4f;

#define ETA_MIN 0.6931471805599453f
#define ETA_MAX 10.0f

__global__ __launch_bounds__(256) void eta_weights_v4(
    const v4f* __restrict__ loss4,
    const float* __restrict__ eta_value,
    v4f* __restrict__ out4,
    int n4)
{
    const int i = blockIdx.x * 256 + threadIdx.x;
    if (i >= n4) return;

    // Uniform address -> s_load; clamp + reciprocal once per wave (hoisted by
    // uniformity analysis), not per element.
    const float eta     = fminf(fmaxf(eta_value[0], ETA_MIN), ETA_MAX);
    const float inv_eta = 1.0f / eta;   // IEEE divide, computed once

    // global_load_b128 with th:TH_LOAD_NT (stream past L2, no reuse)
    v4f l = __builtin_nontemporal_load(loss4 + i);

    v4f w;
#pragma unroll
    for (int k = 0; k < 4; ++k) {
        // v_cmp + v_fmac + v_cndmask per element
        w[k] = (l[k] > eta) ? 0.0f : fmaf(-l[k], inv_eta, 1.0f);
    }

    // global_store_b128 with th:TH_STORE_NT
    __builtin_nontemporal_store(w, out4 + i);
}

// Scalar tail for n % 4 != 0 (not taken for N = 2^25, kept for generality).
__global__ __launch_bounds__(64) void eta_weights_tail(
    const float* __restrict__ loss,
    const float* __restrict__ eta_value,
    float* __restrict__ out,
    int start, int n)
{
    const int i = start + blockIdx.x * 64 + threadIdx.x;
    if (i >= n) return;

    const float eta = fminf(fmaxf(eta_value[0], ETA_MIN), ETA_MAX);
    const float l   = loss[i];
    out[i] = (l > eta) ? 0.0f : (1.0f - l / eta);
}

extern "C" void kernel_launch(void* const* d_in, const int* in_sizes, int n_in,
                              void* d_out, int out_size, void* d_ws, size_t ws_size,
                              hipStream_t stream)
{
    (void)n_in; (void)out_size; (void)d_ws; (void)ws_size;

    const float* loss = (const float*)d_in[0];
    const float* eta  = (const float*)d_in[1];
    float* out        = (float*)d_out;

    const int n   = in_sizes[0];
    const int n4  = n >> 2;           // float4 elements
    const int rem = n - (n4 << 2);

    if (n4 > 0) {
        const int blocks = (n4 + 255) / 256;
        eta_weights_v4<<<blocks, 256, 0, stream>>>(
            (const v4f*)loss, eta, (v4f*)out, n4);
    }
    if (rem > 0) {
        eta_weights_tail<<<1, 64, 0, stream>>>(loss, eta, out, n4 << 2, n);
    }
}